// GaussianBlurDM_53764400611879
// MI455X (gfx1250) — compile-verified
//
#include <hip/hip_runtime.h>
#include <math.h>
#include <stdint.h>

typedef __attribute__((ext_vector_type(2))) float v2f;
typedef __attribute__((ext_vector_type(8))) float v8f;

typedef __attribute__((address_space(1))) int g_i32;   // global
typedef __attribute__((address_space(3))) int l_i32;   // LDS

#define B_   32
#define C_   3
#define H_   256
#define W_   256
#define KS   29
#define PADK 14
#define HID  32
#define NT   1000

#define TIL  16
#define HREG 18              // hidden region per tile (16 + halo 1)
#define ZREG 20              // z region per tile (16 + halo 2)
#define NPIX (HREG*HREG)     // 324
#define NGRP 21              // ceil(324/16)
#define HPAD 336             // padded pixel dim of hidden LDS tile

// ---- async global->LDS staging (CDNA5 GLOBAL_LOAD_ASYNC_TO_LDS_B32) ----
__device__ __forceinline__ void async_ld_f32(const float* g, float* l) {
#if __has_builtin(__builtin_amdgcn_global_load_async_to_lds_b32)
    __builtin_amdgcn_global_load_async_to_lds_b32(
        (g_i32*)(uintptr_t)g, (l_i32*)(uintptr_t)l, 0, 0);
#else
    *l = *g;
#endif
}
__device__ __forceinline__ void async_wait() {
#if __has_builtin(__builtin_amdgcn_s_wait_asynccnt)
    __builtin_amdgcn_s_wait_asynccnt(0);
#elif __has_builtin(__builtin_amdgcn_global_load_async_to_lds_b32)
    asm volatile("s_wait_asynccnt 0" ::: "memory");
#endif
}

// ---------------- kernel 1: per-sample gaussian taps ----------------
__global__ void k_make_kernels(const int* __restrict__ t,
                               const float* __restrict__ sched,
                               float* __restrict__ kcoef) {
    int b = threadIdx.x;
    if (b >= B_) return;
    float sigma = sched[t[b]];
    float w[KS];
    float s = 0.f;
    for (int i = 0; i < KS; ++i) {
        float xg = (float)(i - PADK);
        float e = expf(-0.5f * (xg / sigma) * (xg / sigma));
        w[i] = e; s += e;
    }
    float inv = 1.f / s;
    for (int i = 0; i < KS; ++i) kcoef[b * 32 + i] = w[i] * inv;
}

// ---------------- kernel 2: vertical blur (reflect pad) -------------
__global__ __launch_bounds__(256)
void k_vblur(const float* __restrict__ x, const float* __restrict__ kcoef,
             float* __restrict__ out) {
    int row  = blockIdx.x;            // 0 .. B*C*H-1 (one block per image row)
    int xcol = threadIdx.x;           // 0 .. 255
    int y  = row % H_;
    int bc = row / H_;
    int b  = bc / C_;
    const float* base = x + (size_t)bc * H_ * W_;
    float acc = 0.f;
    for (int j = 0; j < KS; ++j) {
        int yy = y + j - PADK;
        yy = (yy < 0) ? -yy : yy;
        yy = (yy >= H_) ? (2 * H_ - 2 - yy) : yy;
        acc = fmaf(kcoef[b * 32 + j], base[yy * W_ + xcol], acc);
    }
    out[(size_t)row * W_ + xcol] = acc;
}

// ---------------- kernel 3: horizontal blur (reflect pad) -----------
__global__ __launch_bounds__(256)
void k_hblur(const float* __restrict__ tmp, const float* __restrict__ kcoef,
             float* __restrict__ out) {
    __shared__ float line[W_ + KS - 1];   // 284
    int row = blockIdx.x;
    int tid = threadIdx.x;
    int b = row / (C_ * H_);
    const float* base = tmp + (size_t)row * W_;
    for (int i = tid; i < W_ + KS - 1; i += 256) {
        int xx = i - PADK;
        xx = (xx < 0) ? -xx : xx;
        xx = (xx >= W_) ? (2 * W_ - 2 - xx) : xx;
        async_ld_f32(base + xx, &line[i]);
    }
    async_wait();
    __syncthreads();
    float acc = 0.f;
    for (int j = 0; j < KS; ++j)
        acc = fmaf(kcoef[b * 32 + j], line[tid + j], acc);
    out[(size_t)row * W_ + tid] = acc;
}

// ------- kernel 4: conv1 (WMMA f32) + ReLU + conv2 + MSE partials ----
__global__ __launch_bounds__(256)
void k_convloss(const float* __restrict__ z, const float* __restrict__ x,
                const int* __restrict__ t,
                const float* __restrict__ W1, const float* __restrict__ b1,
                const float* __restrict__ tw, const float* __restrict__ W2,
                const float* __restrict__ b2, float* __restrict__ partials) {
    __shared__ float zt[C_][ZREG][ZREG];     // 4.8 KB  z tile + halo 2
    __shared__ float ht[HID][HPAD];          // 43 KB   hidden tile (ch-major)
    __shared__ float wt2[C_ * HID * 9];      // 3.5 KB  conv2 weights
    __shared__ float red[256];

    int x0 = blockIdx.x * TIL, y0 = blockIdx.y * TIL;
    int b  = blockIdx.z;
    int tid = threadIdx.x;

    // Stage z tile async (zero outside image: convs are SAME/zero-pad)
    for (int i = tid; i < C_ * ZREG * ZREG; i += 256) {
        int c = i / (ZREG * ZREG); int r = i % (ZREG * ZREG);
        int zy = r / ZREG, zx = r % ZREG;
        int gy = y0 + zy - 2, gx = x0 + zx - 2;
        if (gy >= 0 && gy < H_ && gx >= 0 && gx < W_)
            async_ld_f32(&z[(((size_t)b * C_ + c) * H_ + gy) * W_ + gx],
                         &zt[c][zy][zx]);
        else
            zt[c][zy][zx] = 0.f;
    }
    for (int i = tid; i < C_ * HID * 9; i += 256)
        async_ld_f32(&W2[i], &wt2[i]);
    async_wait();
    __syncthreads();

    // ---- conv1 as GEMM: [32 x 27] x [27 x Npix] via V_WMMA_F32_16X16X4_F32
    int lane = tid & 31;
    int wv   = tid >> 5;         // wave id 0..7
    int nn   = lane & 15;        // matrix column / A row (M)
    int half = lane >> 4;        // lane half

    // A fragments: W1 flattened [m][k], k = cin*9 + ky*3 + kx (pad 27->28)
    v2f afrag[2][7];
    for (int tile = 0; tile < 2; ++tile)
        for (int s = 0; s < 7; ++s)
            for (int v = 0; v < 2; ++v) {
                int k = 4 * s + 2 * half + v;
                int m = tile * 16 + nn;
                afrag[tile][s][v] = (k < 27) ? W1[m * 27 + k] : 0.f;
            }
    // per-lane bias + time embedding for the channels this lane will write
    float tn = (float)t[b] / (float)NT;
    float emb[2][8];
    for (int tile = 0; tile < 2; ++tile)
        for (int v = 0; v < 8; ++v) {
            int ch = tile * 16 + v + 8 * half;
            emb[tile][v] = b1[ch] + tn * tw[ch];
        }

    for (int g = wv; g < NGRP; g += 8) {           // wave-uniform loop: EXEC all 1s
        int p  = g * 16 + nn;
        int pe = (p < NPIX) ? p : (NPIX - 1);      // clamp addr for tail columns
        int py = pe / HREG, px = pe % HREG;
        v8f acc0 = {}; v8f acc1 = {};
        for (int s = 0; s < 7; ++s) {
            v2f bfrag;
            for (int v = 0; v < 2; ++v) {
                int k  = 4 * s + 2 * half + v;
                int kc = (k < 27) ? k : 26;        // branchless: clamp + select
                int c = kc / 9, r = kc % 9;
                int ky = r / 3, kx = r % 3;
                float ldv = zt[c][py + ky][px + kx];
                bfrag[v] = (k < 27) ? ldv : 0.f;
            }
            acc0 = __builtin_amdgcn_wmma_f32_16x16x4_f32(
                false, afrag[0][s], false, bfrag, (short)0, acc0, false, false);
            acc1 = __builtin_amdgcn_wmma_f32_16x16x4_f32(
                false, afrag[1][s], false, bfrag, (short)0, acc1, false, false);
        }
        // epilogue: +emb, ReLU; zero out-of-image h pixels (conv2 zero-pads)
        int gy = y0 + py - 1, gx = x0 + px - 1;
        float maskf = ((p < NPIX) && gy >= 0 && gy < H_ &&
                       gx >= 0 && gx < W_) ? 1.f : 0.f;
        for (int v = 0; v < 8; ++v) {
            int ch0 = v + 8 * half;                // D layout: M = v + 8*half
            ht[ch0][p]      = fmaxf(acc0[v] + emb[0][v], 0.f) * maskf;
            ht[16 + ch0][p] = fmaxf(acc1[v] + emb[1][v], 0.f) * maskf;
        }
    }
    __syncthreads();

    // ---- conv2 (Cout=3, VALU) + squared error --------------------------
    int oy = tid >> 4, ox = tid & 15;
    int gy = y0 + oy, gx = x0 + ox;
    float local = 0.f;
    for (int co = 0; co < C_; ++co) {
        float sum = b2[co];
        for (int cin = 0; cin < HID; ++cin) {
            const float* hrow = &ht[cin][0];
            const float* wr   = &wt2[(co * HID + cin) * 9];
            for (int dy = 0; dy < 3; ++dy)
                for (int dx = 0; dx < 3; ++dx)
                    sum = fmaf(wr[dy * 3 + dx],
                               hrow[(oy + dy) * HREG + (ox + dx)], sum);
        }
        float err = x[(((size_t)b * C_ + co) * H_ + gy) * W_ + gx] - sum;
        local = fmaf(err, err, local);
    }
    red[tid] = local;
    __syncthreads();
    for (int sft = 128; sft > 0; sft >>= 1) {       // fixed-order tree: deterministic
        if (tid < sft) red[tid] += red[tid + sft];
        __syncthreads();
    }
    if (tid == 0)
        partials[((size_t)blockIdx.z * gridDim.y + blockIdx.y) * gridDim.x
                 + blockIdx.x] = red[0];
}

// ---------------- kernel 5: final deterministic reduce ----------------
__global__ void k_reduce(const float* __restrict__ partials, int n,
                         float* __restrict__ out) {
    __shared__ double red[256];
    int tid = threadIdx.x;
    double s = 0.0;
    for (int i = tid; i < n; i += 256) s += (double)partials[i];
    red[tid] = s;
    __syncthreads();
    for (int sft = 128; sft > 0; sft >>= 1) {
        if (tid < sft) red[tid] += red[tid + sft];
        __syncthreads();
    }
    if (tid == 0)
        out[0] = (float)(red[0] / (double)((size_t)B_ * C_ * H_ * W_));
}

extern "C" void kernel_launch(void* const* d_in, const int* in_sizes, int n_in,
                              void* d_out, int out_size, void* d_ws, size_t ws_size,
                              hipStream_t stream) {
    (void)in_sizes; (void)n_in; (void)out_size; (void)ws_size;
    const float* x     = (const float*)d_in[0];
    const int*   t     = (const int*)  d_in[1];
    const float* W1    = (const float*)d_in[2];
    const float* b1    = (const float*)d_in[3];
    const float* tw    = (const float*)d_in[4];
    const float* W2    = (const float*)d_in[5];
    const float* b2    = (const float*)d_in[6];
    const float* sched = (const float*)d_in[7];
    float* out = (float*)d_out;

    char* ws = (char*)d_ws;
    const size_t imgBytes = (size_t)B_ * C_ * H_ * W_ * sizeof(float); // 24 MB
    float* kcoef    = (float*)ws;                       // 32*32 f32
    float* partials = (float*)(ws + 4096);              // 8192 f32
    float* tmp      = (float*)(ws + 65536);             // vertical-blur temp
    float* zbuf     = (float*)(ws + 65536 + imgBytes);  // blurred z

    k_make_kernels<<<1, 32, 0, stream>>>(t, sched, kcoef);
    k_vblur<<<B_ * C_ * H_, W_, 0, stream>>>(x, kcoef, tmp);
    k_hblur<<<B_ * C_ * H_, W_, 0, stream>>>(tmp, kcoef, zbuf);
    dim3 grid(W_ / TIL, H_ / TIL, B_);
    k_convloss<<<grid, 256, 0, stream>>>(zbuf, x, t, W1, b1, tw, W2, b2, partials);
    k_reduce<<<1, 256, 0, stream>>>(partials, (W_ / TIL) * (H_ / TIL) * B_, out);
}